// CoverTreeLoss_81312320848190
// MI455X (gfx1250) — compile-verified
//
#include <hip/hip_runtime.h>
#include <hip/hip_bf16.h>

// Problem constants (match reference)
constexpr int kC = 100000;   // classes
constexpr int kL = 120000;   // weight rows (classes + internal nodes)
constexpr int kD = 128;      // feature dim
constexpr int kDEPTH = 8;    // tree path length
constexpr int kB = 2048;     // batch

typedef __attribute__((ext_vector_type(2))) float v2f;
typedef __attribute__((ext_vector_type(8))) float v8f;

// ---------------------------------------------------------------------------
// Kernel 1: added_weights[c, :] = sum_j weights[path_idx[c, j], :]
// One wave (32 lanes) per class; each lane handles a float4 (128 floats/row).
// ---------------------------------------------------------------------------
__global__ __launch_bounds__(256) void aw_kernel(const float* __restrict__ w,
                                                 const int* __restrict__ path,
                                                 float* __restrict__ aw) {
    const int c = blockIdx.x * (blockDim.x >> 5) + (threadIdx.x >> 5);
    const int lane = threadIdx.x & 31;
    if (c >= kC) return;
    const int* p = path + (size_t)c * kDEPTH;
    float4 acc = make_float4(0.f, 0.f, 0.f, 0.f);
#pragma unroll
    for (int j = 0; j < kDEPTH; ++j) {
        const float4* row = (const float4*)(w + (size_t)p[j] * kD);
        float4 v = row[lane];
        acc.x += v.x; acc.y += v.y; acc.z += v.z; acc.w += v.w;
    }
    ((float4*)(aw + (size_t)c * kD))[lane] = acc;
}

// ---------------------------------------------------------------------------
// Kernel 2: logits[m, n] = sum_k x[m, k] * aw[n, k]   (f32 WMMA 16x16x4)
// Block tile: M=128 x N=80, K=128 fully staged in LDS via ASYNC global->LDS.
// 4 waves; wave w owns M bands [16w,16w+16) and [64+16w,64+16w+16), and all
// five N-subtiles of 16 -> acc[2][5] (10 independent WMMA chains).
// LDS stride 132 floats (pad 4) keeps fragment reads bank-conflict-free.
// Fragment layout (ISA 7.12.2, 32-bit A 16x4 / B 4x16): lane -> (t=lane%16,
// h=lane/16); per k-step k0 both A and B fragments are the contiguous float2
// at [row16 * stride + k0 + 2*h].
// ---------------------------------------------------------------------------
__global__ __launch_bounds__(128) void gemm_kernel(const float* __restrict__ x,
                                                   const float* __restrict__ aw,
                                                   float* __restrict__ logits) {
    constexpr int LDSS = kD + 4;  // 132
    __shared__ float lds[(128 + 80) * LDSS];
    float* As = lds;               // 128 x 132
    float* Bs = lds + 128 * LDSS;  // 80 x 132

    const int n0 = blockIdx.x * 80;
    const int m0 = blockIdx.y * 128;
    const int tid = threadIdx.x;

    // Stage A tile (128x128 f32): 4096 x 16B chunks, 32 per thread, async.
#pragma unroll
    for (int i = 0; i < 32; ++i) {
        int f = tid + 128 * i;
        int row = f >> 5, c4 = f & 31;
        const float* g = x + (size_t)(m0 + row) * kD + c4 * 4;
        unsigned ldsa = (unsigned)(uintptr_t)&As[row * LDSS + c4 * 4];
        asm volatile("global_load_async_to_lds_b128 %0, %1, off"
                     :: "v"(ldsa), "v"(g) : "memory");
    }
    // Stage B tile (80x128 f32): 2560 x 16B chunks, 20 per thread, async.
#pragma unroll
    for (int i = 0; i < 20; ++i) {
        int f = tid + 128 * i;
        int row = f >> 5, c4 = f & 31;
        const float* g = aw + (size_t)(n0 + row) * kD + c4 * 4;
        unsigned ldsa = (unsigned)(uintptr_t)&Bs[row * LDSS + c4 * 4];
        asm volatile("global_load_async_to_lds_b128 %0, %1, off"
                     :: "v"(ldsa), "v"(g) : "memory");
    }
    asm volatile("s_wait_asynccnt 0x0" ::: "memory");
    __syncthreads();

    const int w = tid >> 5;
    const int lane = tid & 31;
    const int t16 = lane & 15;
    const int half = lane >> 4;

    v8f acc[2][5];
    const v8f zero = {0.f, 0.f, 0.f, 0.f, 0.f, 0.f, 0.f, 0.f};
#pragma unroll
    for (int mi = 0; mi < 2; ++mi)
#pragma unroll
        for (int j = 0; j < 5; ++j) acc[mi][j] = zero;

    const float* A0 = &As[(16 * w + t16) * LDSS + 2 * half];
    const float* A1 = A0 + 64 * LDSS;
    const float* B0 = &Bs[t16 * LDSS + 2 * half];

#pragma unroll 2
    for (int k0 = 0; k0 < kD; k0 += 4) {
        v2f a0 = *(const v2f*)(A0 + k0);
        v2f a1 = *(const v2f*)(A1 + k0);
#pragma unroll
        for (int j = 0; j < 5; ++j) {
            v2f b = *(const v2f*)(B0 + j * 16 * LDSS + k0);
            acc[0][j] = __builtin_amdgcn_wmma_f32_16x16x4_f32(
                false, a0, false, b, (short)0, acc[0][j], false, false);
            acc[1][j] = __builtin_amdgcn_wmma_f32_16x16x4_f32(
                false, a1, false, b, (short)0, acc[1][j], false, false);
        }
    }

    // D layout: VGPR r -> M = r (lanes 0-15) / r+8 (lanes 16-31), N = lane%16.
#pragma unroll
    for (int mi = 0; mi < 2; ++mi) {
        const int mbase = m0 + 64 * mi + 16 * w + 8 * half;
#pragma unroll
        for (int j = 0; j < 5; ++j) {
            const int col = n0 + 16 * j + t16;
#pragma unroll
            for (int r = 0; r < 8; ++r) {
                logits[(size_t)(mbase + r) * kC + col] = acc[mi][j][r];
            }
        }
    }
}

// ---------------------------------------------------------------------------
// Kernel 3: per-row streaming logsumexp + target logit.
// rowterm[b] = logits[b, y[b]] - logsumexp_c(logits[b, c])
// ---------------------------------------------------------------------------
__global__ __launch_bounds__(256) void lse_kernel(const float* __restrict__ logits,
                                                  const int* __restrict__ y,
                                                  float* __restrict__ rowterm) {
    const int b = blockIdx.x;
    const float* row = logits + (size_t)b * kC;

    float m = -3.402823466e38f, s = 0.f;
    for (int c = threadIdx.x; c < kC; c += 256) {
        float v = row[c];
        if (v <= m) {
            s += expf(v - m);
        } else {
            s = s * expf(m - v) + 1.f;
            m = v;
        }
    }

    __shared__ float sm[256], ss[256];
    sm[threadIdx.x] = m;
    ss[threadIdx.x] = s;
    __syncthreads();
    for (int off = 128; off > 0; off >>= 1) {
        if (threadIdx.x < off) {
            float m1 = sm[threadIdx.x], s1 = ss[threadIdx.x];
            float m2 = sm[threadIdx.x + off], s2 = ss[threadIdx.x + off];
            float M = fmaxf(m1, m2);
            ss[threadIdx.x] = s1 * expf(m1 - M) + s2 * expf(m2 - M);
            sm[threadIdx.x] = M;
        }
        __syncthreads();
    }
    if (threadIdx.x == 0) {
        float lse = sm[0] + logf(ss[0]);
        rowterm[b] = row[y[b]] - lse;
    }
}

// ---------------------------------------------------------------------------
// Kernel 4: loss = -mean(rowterm). Fixed-order reduction -> deterministic.
// ---------------------------------------------------------------------------
__global__ __launch_bounds__(256) void loss_kernel(const float* __restrict__ rowterm,
                                                   float* __restrict__ out) {
    __shared__ float sh[256];
    float a = 0.f;
    for (int i = threadIdx.x; i < kB; i += 256) a += rowterm[i];
    sh[threadIdx.x] = a;
    __syncthreads();
    for (int off = 128; off > 0; off >>= 1) {
        if (threadIdx.x < off) sh[threadIdx.x] += sh[threadIdx.x + off];
        __syncthreads();
    }
    if (threadIdx.x == 0) out[0] = -sh[0] / (float)kB;
}

// ---------------------------------------------------------------------------
extern "C" void kernel_launch(void* const* d_in, const int* in_sizes, int n_in,
                              void* d_out, int out_size, void* d_ws, size_t ws_size,
                              hipStream_t stream) {
    const float* x       = (const float*)d_in[0];   // [B, D]
    const int*   y       = (const int*)d_in[1];     // [B]
    const float* weights = (const float*)d_in[2];   // [L, D]
    const int*   path    = (const int*)d_in[3];     // [C, DEPTH]

    float* out     = (float*)d_out;                 // [0]=loss, [1:]=logits [B, C]
    float* logits  = out + 1;
    float* aw      = (float*)d_ws;                  // [C, D]
    float* rowterm = aw + (size_t)kC * kD;          // [B]

    // 1) gather-sum of tree-path rows
    aw_kernel<<<kC / 8, 256, 0, stream>>>(weights, path, aw);

    // 2) WMMA f32 GEMM: [B, D] x [C, D]^T -> [B, C]
    dim3 grid(kC / 80, kB / 128);
    gemm_kernel<<<grid, 128, 0, stream>>>(x, aw, logits);

    // 3) per-row logsumexp + target term
    lse_kernel<<<kB, 256, 0, stream>>>(logits, y, rowterm);

    // 4) final deterministic mean reduction
    loss_kernel<<<1, 256, 0, stream>>>(rowterm, out);
}